// GraphDocEncoder_84464826843205
// MI455X (gfx1250) — compile-verified
//
#include <hip/hip_runtime.h>
#include <hip/hip_bf16.h>

// ---------------------------------------------------------------------------
// GraphDocEncoder attention — MI455X (gfx1250, wave32)
//
// Key insight: the "local" mask restricts softmax support to the <=32 keys in
// rel_bbox_index (all other positions hit exp(-1e8-max)==0 in fp32), so the
// attention is T=32-sparse: no LxL score matrix, and rel_pos/rel_2d_pos are
// gathered only at the selected keys. GEMMs run on V_WMMA_F32_16X16X4_F32
// (full fp32 precision). Dominant HBM traffic is the single 403 MB pass over
// rel_bbox_emb, streamed with b128 loads (+prefetch) at ~23.3 TB/s.
// ---------------------------------------------------------------------------

typedef __attribute__((ext_vector_type(2))) float v2f;
typedef __attribute__((ext_vector_type(8))) float v8f;

#define Bc 4
#define Lc 1024
#define Tc 32
#define Dc 768
#define Hc 12
#define DHc 64
#define SCALEc 0.125f   // 1/sqrt(64)

// ---------------------------------------------------------------------------
// Kernel 1: fp32 WMMA GEMM   C[M,N] = A[M,K] @ B[K,N] + bias[N]
// block = 256 threads = 8 waves stacked in M; each wave owns a 16x64 tile
// (4 accumulators, A fragment reused across the 4 N-subtiles).
// Block tile: 128M x 64N.  Requires M%128==0, N%64==0, K%4==0.
// No divergence anywhere (EXEC all 1s for WMMA).
// ---------------------------------------------------------------------------
__global__ __launch_bounds__(256) void gemm_f32_wmma(
    const float* __restrict__ A, const float* __restrict__ Bm,
    const float* __restrict__ bias, float* __restrict__ C,
    int M, int N, int K)
{
    const int lane  = threadIdx.x & 31;
    const int wave  = threadIdx.x >> 5;              // 0..7, M direction
    const int tileM = blockIdx.y * 128 + wave * 16;
    const int tileN = blockIdx.x * 64;

    const int half = lane >> 4;                      // 0: lanes 0-15, 1: 16-31
    const int l16  = lane & 15;
    const int arow = tileM + l16;                    // A row owned by this lane

    v8f acc0 = {}, acc1 = {}, acc2 = {}, acc3 = {};

    // A fragment source: lane<16 -> K {k,k+1}; lane>=16 -> K {k+2,k+3}
    const float* aBase = A + (size_t)arow * K + half * 2;
    // B fragment source: vgpr0 = row {k | k+2}, vgpr1 = row {k+1 | k+3}
    const float* bBase = Bm + (size_t)(half * 2) * N + tileN + l16;

    for (int k = 0; k < K; k += 4) {
        const v2f a = *reinterpret_cast<const v2f*>(aBase + k);   // b64 load
        const float* bp = bBase + (size_t)k * N;
        v2f b0, b1, b2, b3;
        b0.x = bp[0];   b0.y = bp[N];
        b1.x = bp[16];  b1.y = bp[N + 16];
        b2.x = bp[32];  b2.y = bp[N + 32];
        b3.x = bp[48];  b3.y = bp[N + 48];
        acc0 = __builtin_amdgcn_wmma_f32_16x16x4_f32(false, a, false, b0, (short)0, acc0, false, false);
        acc1 = __builtin_amdgcn_wmma_f32_16x16x4_f32(false, a, false, b1, (short)0, acc1, false, false);
        acc2 = __builtin_amdgcn_wmma_f32_16x16x4_f32(false, a, false, b2, (short)0, acc2, false, false);
        acc3 = __builtin_amdgcn_wmma_f32_16x16x4_f32(false, a, false, b3, (short)0, acc3, false, false);
    }

    // C/D layout: lane<16 holds rows tileM+0..7 (vgpr r -> row r),
    //             lane>=16 holds rows tileM+8..15; col = subtile base + l16.
    float* cRow = C + (size_t)(tileM + half * 8) * N + tileN + l16;
#pragma unroll
    for (int c = 0; c < 4; ++c) {
        const v8f acc = (c == 0) ? acc0 : (c == 1) ? acc1 : (c == 2) ? acc2 : acc3;
        const int col = c * 16;
        const float bv = bias ? bias[tileN + col + l16] : 0.0f;
#pragma unroll
        for (int r = 0; r < 8; ++r)
            cRow[(size_t)r * N + col] = acc[r] + bv;
    }
}

// ---------------------------------------------------------------------------
// Kernel 2: fused bias  bqrb[d] = sum_e bq[e]*Wrb[e,d] + brb[d]
// ---------------------------------------------------------------------------
__global__ __launch_bounds__(256) void bias_fuse(
    const float* __restrict__ bq, const float* __restrict__ Wrb,
    const float* __restrict__ brb, float* __restrict__ bqrb, int Dd)
{
    int d = blockIdx.x * blockDim.x + threadIdx.x;
    if (d >= Dd) return;
    float sum = brb[d];
    for (int e = 0; e < Dd; ++e)
        sum += bq[e] * Wrb[(size_t)e * Dd + d];
    bqrb[d] = sum;
}

// ---------------------------------------------------------------------------
// Kernel 3: bbox scores.  One wave per (b,i,t).
//   bbox[(b,i,t),h] = scale * dot(qb[b,i, h*64:(h+1)*64], rbe[b,i,t, h*64:..])
// Streams the 403 MB rel_bbox_emb with b128 loads (HBM-rate bound at 23 TB/s).
// Each float4 chunk lies inside one head; iteration m covers heads 2m (lanes
// 0-15) and 2m+1 (lanes 16-31), reduced with 16-lane butterflies.
// ---------------------------------------------------------------------------
__global__ __launch_bounds__(256) void bbox_scores(
    const float* __restrict__ qb, const float* __restrict__ rbe,
    float* __restrict__ bbox)
{
    const int waveId = (int)((blockIdx.x * blockDim.x + threadIdx.x) >> 5);
    const int lane   = threadIdx.x & 31;
    if (waveId >= Bc * Lc * Tc) return;

    const int bi = waveId / Tc;                       // b*L + i
    const float4* r = reinterpret_cast<const float4*>(rbe + (size_t)waveId * Dc);
    const float4* q = reinterpret_cast<const float4*>(qb  + (size_t)bi * Dc);
    float* o = bbox + (size_t)waveId * Hc;

    // keep the HBM stream ahead: prefetch next t-row (gfx1250 global_prefetch)
    __builtin_prefetch(reinterpret_cast<const char*>(r) + Dc * sizeof(float) +
                       lane * 16, 0, 1);

    const int sub = lane >> 4;                        // which head of the pair
#pragma unroll
    for (int m = 0; m < 6; ++m) {
        const int f4 = m * 32 + lane;                 // float4 index 0..191
        const float4 rv = r[f4];
        const float4 qv = q[f4];
        float p = rv.x * qv.x + rv.y * qv.y + rv.z * qv.z + rv.w * qv.w;
#pragma unroll
        for (int off = 8; off; off >>= 1)             // 16-lane segmented sum
            p += __shfl_xor(p, off, 32);
        if ((lane & 15) == 0) o[2 * m + sub] = p * SCALEc;
    }
}

// ---------------------------------------------------------------------------
// Kernel 4: sparse attention.  One wave per (b,h,i); lane = t in 0..31.
//   S_t = scale*(q_i . k_{j_t}) + sum_{t':j_t'==j_t} bbox_t' + rp + rp2
//   (only first occurrence of each j survives; duplicates -> excluded)
//   softmax over 32 lanes, then lane d accumulates ctx[d], ctx[d+32].
// ---------------------------------------------------------------------------
__global__ __launch_bounds__(256) void attn(
    const float* __restrict__ q, const float* __restrict__ k,
    const float* __restrict__ v, const int* __restrict__ idx,
    const float* __restrict__ bbox, const float* __restrict__ relpos,
    const float* __restrict__ rel2d, const unsigned char* __restrict__ amask,
    float* __restrict__ out)
{
    const int waveId = (int)((blockIdx.x * blockDim.x + threadIdx.x) >> 5);
    const int lane   = threadIdx.x & 31;                // = t
    if (waveId >= Bc * Hc * Lc) return;

    const int i  = waveId % Lc;
    const int bh = waveId / Lc;
    const int h  = bh % Hc;
    const int b  = bh / Hc;

    const int j = idx[((size_t)b * Lc + i) * Tc + lane];

    // scaled q.k over the head slice (b128 loads; k rows L2-resident)
    const float* qrow = q + ((size_t)b * Lc + i) * Dc + h * DHc;
    const float* krow = k + ((size_t)b * Lc + j) * Dc + h * DHc;
    float s = 0.f;
#pragma unroll
    for (int d = 0; d < DHc; d += 4) {
        const float4 kv = *reinterpret_cast<const float4*>(krow + d);
        const float4 qv = *reinterpret_cast<const float4*>(qrow + d);
        s += qv.x * kv.x + qv.y * kv.y + qv.z * kv.z + qv.w * kv.w;
    }
    s *= SCALEc;

    const float bb = bbox[(((size_t)b * Lc + i) * Tc + lane) * Hc + h];

    // dedup: sum bbox over equal indices; only the first occurrence survives
    float bbsum = 0.f;
    bool  first = true;
#pragma unroll
    for (int tt = 0; tt < 32; ++tt) {
        const int   jt = __shfl(j,  tt, 32);
        const float bt = __shfl(bb, tt, 32);
        if (jt == j) {
            bbsum += bt;
            if (tt < lane) first = false;
        }
    }

    const size_t rpo = ((size_t)h * Lc + i) * Lc + j;
    float S = s + bbsum + relpos[rpo] + rel2d[rpo];
    if (amask[(size_t)b * Lc + j]) S = -1e30f;   // attention_mask (all-false here)
    if (!first) S = -1e30f;                      // duplicate -> excluded

    // 32-lane softmax
    float m = S;
#pragma unroll
    for (int off = 16; off; off >>= 1) m = fmaxf(m, __shfl_xor(m, off, 32));
    float e = __expf(S - m);                     // -1e30 lanes underflow to 0
    float sum = e;
#pragma unroll
    for (int off = 16; off; off >>= 1) sum += __shfl_xor(sum, off, 32);
    const float p = e / sum;

    // context: lane d accumulates out[d], out[d+32] with coalesced v reads
    float c0 = 0.f, c1 = 0.f;
    for (int tt = 0; tt < 32; ++tt) {
        const float pt = __shfl(p, tt, 32);
        const int   jt = __shfl(j, tt, 32);
        const float* vrow = v + ((size_t)b * Lc + jt) * Dc + h * DHc;
        c0 += pt * vrow[lane];
        c1 += pt * vrow[lane + 32];
    }
    float* orow = out + ((size_t)b * Lc + i) * Dc + h * DHc;
    orow[lane]      = c0;
    orow[lane + 32] = c1;
}

// ---------------------------------------------------------------------------
extern "C" void kernel_launch(void* const* d_in, const int* in_sizes, int n_in,
                              void* d_out, int out_size, void* d_ws, size_t ws_size,
                              hipStream_t stream)
{
    const float*         hs     = (const float*)d_in[0];   // (B,L,D)
    const float*         rbe    = (const float*)d_in[1];   // (B,L,T,D)
    const int*           idx    = (const int*)d_in[2];     // (B,L,T)
    const unsigned char* amask  = (const unsigned char*)d_in[3]; // (B,1,1,L) bool
    const float*         relpos = (const float*)d_in[4];   // (1,H,L,L)
    const float*         rel2d  = (const float*)d_in[5];   // (1,H,L,L)
    const float*         Wq     = (const float*)d_in[6];
    const float*         bq     = (const float*)d_in[7];
    const float*         Wk     = (const float*)d_in[8];
    const float*         bk     = (const float*)d_in[9];
    const float*         Wv     = (const float*)d_in[10];
    const float*         bv     = (const float*)d_in[11];
    const float*         Wrb    = (const float*)d_in[12];
    const float*         brb    = (const float*)d_in[13];

    float* out = (float*)d_out;

    // workspace layout (floats)
    const size_t nBL   = (size_t)Bc * Lc;        // 4096
    const size_t nQKV  = nBL * Dc;               // 3,145,728
    float* ws    = (float*)d_ws;
    float* qbuf  = ws;                           // q
    float* kbuf  = qbuf + nQKV;                  // k
    float* vbuf  = kbuf + nQKV;                  // v
    float* qbbuf = vbuf + nQKV;                  // q @ Wrb (q_bbox, unscaled)
    float* Wqrb  = qbbuf + nQKV;                 // fused Wq@Wrb   (D*D)
    float* bqrb  = Wqrb + (size_t)Dc * Dc;       // fused bias     (D)
    float* bbox  = bqrb + Dc;                    // (B,L,T,H)

    const int M = (int)nBL;   // 4096
    const int N = Dc;         // 768
    const int K = Dc;         // 768

    // 0) fuse Wqrb = Wq @ Wrb ; bqrb = bq @ Wrb + brb
    gemm_f32_wmma<<<dim3(Dc / 64, Dc / 128), 256, 0, stream>>>(
        Wq, Wrb, nullptr, Wqrb, Dc, Dc, Dc);
    bias_fuse<<<(Dc + 255) / 256, 256, 0, stream>>>(bq, Wrb, brb, bqrb, Dc);

    // 1) projections: q, k, v, q_bbox — fp32 WMMA GEMMs (block tile 128x64)
    gemm_f32_wmma<<<dim3(N / 64, M / 128), 256, 0, stream>>>(hs, Wq,   bq,   qbuf,  M, N, K);
    gemm_f32_wmma<<<dim3(N / 64, M / 128), 256, 0, stream>>>(hs, Wk,   bk,   kbuf,  M, N, K);
    gemm_f32_wmma<<<dim3(N / 64, M / 128), 256, 0, stream>>>(hs, Wv,   bv,   vbuf,  M, N, K);
    gemm_f32_wmma<<<dim3(N / 64, M / 128), 256, 0, stream>>>(hs, Wqrb, bqrb, qbbuf, M, N, K);

    // 2) bbox scores: one wave per (b,i,t) — streams rel_bbox_emb (403 MB)
    {
        const long long waves = (long long)Bc * Lc * Tc;        // 131072
        const long long thr   = waves * 32;
        bbox_scores<<<(int)((thr + 255) / 256), 256, 0, stream>>>(qbbuf, rbe, bbox);
    }

    // 3) sparse attention: one wave per (b,h,i)
    {
        const long long waves = (long long)Bc * Hc * Lc;        // 49152
        const long long thr   = waves * 32;
        attn<<<(int)((thr + 255) / 256), 256, 0, stream>>>(
            qbuf, kbuf, vbuf, idx, bbox, relpos, rel2d, amask, out);
    }
}